// GNNLayer_26096221290519
// MI455X (gfx1250) — compile-verified
//
#include <hip/hip_runtime.h>
#include <hip/hip_bf16.h>
#include <math.h>

typedef __attribute__((ext_vector_type(2))) float v2f;
typedef __attribute__((ext_vector_type(8))) float v8f;

// ---------------------------------------------------------------------------
// zero scratch accumulator (segment-sum target) each call
// ---------------------------------------------------------------------------
__global__ void zero_f32(float* __restrict__ p, long n) {
  long i = (long)blockIdx.x * blockDim.x + threadIdx.x;
  const long stride = (long)gridDim.x * blockDim.x;
  for (; i < n; i += stride) p[i] = 0.0f;
}

// ---------------------------------------------------------------------------
// C[r, 0:64] = A[(gather? gather[r] : r), 0:64] @ B[64x64]  (+ bias[0:64])
// One wave32 computes a 16-row x 64-col strip with V_WMMA_F32_16X16X4_F32.
// A fragment (16x4 f32): lanes 0-15 hold K=0,1 ; lanes 16-31 hold K=2,3.
// B fragment (4x16 f32): same K split across lane halves, N = lane%16.
// B is staged in LDS PRE-SWIZZLED into fragment layout so each inner-loop
// fragment load is one ds_load_b64 into an even-aligned VGPR pair:
//   Bs[((kk*4 + nt)*32 + lane)*2 + {0,1}] = B[4kk + 2kh + {0,1}][16nt + mr]
// D (16x16 f32, 8 VGPRs): elem v -> row v (lanes 0-15) / row v+8 (lanes 16-31).
// ---------------------------------------------------------------------------
__global__ void gemm64_wmma(const float* __restrict__ A,
                            const int*   __restrict__ gather,
                            const float* __restrict__ B,
                            const float* __restrict__ bias,
                            float*       __restrict__ C,
                            int M) {
  __shared__ float Bs[16 * 4 * 32 * 2];   // 4096 floats = 16 KB
  for (int idx = threadIdx.x; idx < 16 * 4 * 32; idx += blockDim.x) {
    const int kk  = idx >> 7;          // k-step  (0..15)
    const int nt  = (idx >> 5) & 3;    // n-tile  (0..3)
    const int ln  = idx & 31;          // lane    (0..31)
    const int kh  = ln >> 4;
    const int mr  = ln & 15;
    const int row = 4 * kk + 2 * kh;
    const int col = 16 * nt + mr;
    Bs[idx * 2 + 0] = B[(row + 0) * 64 + col];
    Bs[idx * 2 + 1] = B[(row + 1) * 64 + col];
  }
  __syncthreads();

  const int lane = threadIdx.x & 31;
  const int wave = threadIdx.x >> 5;
  const int tile = blockIdx.x * (blockDim.x >> 5) + wave;
  const int row0 = tile * 16;
  if (row0 >= M) return;

  const int mr = lane & 15;   // A row within tile / D column within n-tile
  const int kh = lane >> 4;   // which K-pair this lane half holds

  int r  = row0 + mr;
  int ar = (r < M) ? r : (M - 1);          // clamp OOB rows (not stored later)
  if (gather) ar = gather[ar];
  const float* __restrict__ Arow = A + (long)ar * 64;
  const float* __restrict__ Bf   = Bs + lane * 2;   // this lane's fragment slot

  v8f acc0 = {}, acc1 = {}, acc2 = {}, acc3 = {};
#pragma unroll
  for (int kk = 0; kk < 16; ++kk) {
    const v2f a  = *(const v2f*)(Arow + kk * 4 + 2 * kh);      // global_load_b64
    const v2f b0 = *(const v2f*)(Bf + (kk * 128 +  0) );       // ds_load_b64
    const v2f b1 = *(const v2f*)(Bf + (kk * 128 + 64) );
    const v2f b2 = *(const v2f*)(Bf + (kk * 128 + 128));
    const v2f b3 = *(const v2f*)(Bf + (kk * 128 + 192));
    acc0 = __builtin_amdgcn_wmma_f32_16x16x4_f32(false, a, false, b0, (short)0, acc0, false, false);
    acc1 = __builtin_amdgcn_wmma_f32_16x16x4_f32(false, a, false, b1, (short)0, acc1, false, false);
    acc2 = __builtin_amdgcn_wmma_f32_16x16x4_f32(false, a, false, b2, (short)0, acc2, false, false);
    acc3 = __builtin_amdgcn_wmma_f32_16x16x4_f32(false, a, false, b3, (short)0, acc3, false, false);
  }

  const float bv0 = bias ? bias[mr +  0] : 0.0f;
  const float bv1 = bias ? bias[mr + 16] : 0.0f;
  const float bv2 = bias ? bias[mr + 32] : 0.0f;
  const float bv3 = bias ? bias[mr + 48] : 0.0f;

#pragma unroll
  for (int v = 0; v < 8; ++v) {
    const int orow = row0 + v + 8 * kh;
    if (orow < M) {
      float* __restrict__ Crow = C + (long)orow * 64;
      Crow[mr +  0] = acc0[v] + bv0;
      Crow[mr + 16] = acc1[v] + bv1;
      Crow[mr + 32] = acc2[v] + bv2;
      Crow[mr + 48] = acc3[v] + bv3;
    }
  }
}

// ---------------------------------------------------------------------------
// Per-edge streaming pass: one wave32 per edge, 2 f32 per lane (64-dim).
// attn = relu(hproj[sub] + rproj[rel] + qattn[r_idx])   (all precomputed GEMMs)
// alpha_temp = attn . w_alpha_w + b ; alpha = sigmoid
// message = alpha * (hidden[sub] + rela[rel]) ; atomic segment-sum into agg.
// ---------------------------------------------------------------------------
__global__ void edge_kernel(const int*   __restrict__ edges,
                            const float* __restrict__ hidden,
                            const float* __restrict__ rela,
                            const float* __restrict__ hproj,
                            const float* __restrict__ rproj,
                            const float* __restrict__ qattn,
                            const float* __restrict__ w_alpha_w,
                            const float* __restrict__ w_alpha_b,
                            float* __restrict__ out_alpha,
                            float* __restrict__ out_msg,
                            float* __restrict__ out_obj,
                            float* __restrict__ out_atemp,
                            float* __restrict__ agg,
                            int n_edge) {
  const int  lane = threadIdx.x & 31;
  const int  wave = threadIdx.x >> 5;
  const long e    = (long)blockIdx.x * (blockDim.x >> 5) + wave;
  if (e >= (long)n_edge) return;

  const long eb    = e * 6;
  const int  r_idx = edges[eb + 0];
  const int  rel   = edges[eb + 2];
  const int  sub   = edges[eb + 4];
  const int  obj   = edges[eb + 5];

  const int c = lane * 2;
  const v2f hs = *(const v2f*)(hidden + (long)sub   * 64 + c);
  const v2f hr = *(const v2f*)(rela   + (long)rel   * 64 + c);
  const v2f hp = *(const v2f*)(hproj  + (long)sub   * 64 + c);
  const v2f rp = *(const v2f*)(rproj  + (long)rel   * 64 + c);
  const v2f qa = *(const v2f*)(qattn  + (long)r_idx * 64 + c);
  const v2f wa = *(const v2f*)(w_alpha_w + c);

  float ax = hp.x + rp.x + qa.x; ax = ax > 0.0f ? ax : 0.0f;
  float ay = hp.y + rp.y + qa.y; ay = ay > 0.0f ? ay : 0.0f;
  float part = ax * wa.x + ay * wa.y;
#pragma unroll
  for (int off = 16; off > 0; off >>= 1)
    part += __shfl_xor(part, off, 32);

  const float atemp = part + w_alpha_b[0];
  const float alpha = 1.0f / (1.0f + __expf(-atemp));

  v2f m;
  m.x = alpha * (hs.x + hr.x);
  m.y = alpha * (hs.y + hr.y);
  *(v2f*)(out_msg + e * 64 + c) = m;

  float* aggp = agg + (long)obj * 64 + c;
  unsafeAtomicAdd(aggp + 0, m.x);   // global_atomic_add_f32 (L2-resident)
  unsafeAtomicAdd(aggp + 1, m.y);

  if (lane == 0) {
    out_alpha[e] = alpha;
    out_atemp[e] = atemp;
    out_obj[e]   = (float)obj;
  }
}

// ---------------------------------------------------------------------------
// Inputs (setup_inputs order):
// 0 q_sub[256] 1 q_rel[256] 2 hidden[100000x64] 3 edges[1000000x6] 4 n_node
// 5 rela_embed[401x64] 6 Ws 7 Wr 8 Wqr [64x64] 9 Wqr_b[64]
// 10 w_alpha_w[64] 11 w_alpha_b[1] 12 W_h[64x64]
// Outputs flat: hidden_new[100000x64] | alpha[1M] | message[1Mx64] | obj[1M] | alpha_temp[1M]
// ---------------------------------------------------------------------------
extern "C" void kernel_launch(void* const* d_in, const int* in_sizes, int n_in,
                              void* d_out, int out_size, void* d_ws, size_t ws_size,
                              hipStream_t stream) {
  const int*   q_rel  = (const int*)  d_in[1];
  const float* hidden = (const float*)d_in[2];
  const int*   edges  = (const int*)  d_in[3];
  const float* rela   = (const float*)d_in[5];
  const float* Ws     = (const float*)d_in[6];
  const float* Wr     = (const float*)d_in[7];
  const float* Wqr    = (const float*)d_in[8];
  const float* Wqr_b  = (const float*)d_in[9];
  const float* waw    = (const float*)d_in[10];
  const float* wab    = (const float*)d_in[11];
  const float* W_h    = (const float*)d_in[12];

  const int n_node  = in_sizes[2] / 64;
  const int n_edge  = in_sizes[3] / 6;
  const int n_query = in_sizes[1];
  const int n_rel   = in_sizes[5] / 64;

  float* out        = (float*)d_out;
  float* out_hidden = out;                                // n_node*64
  float* out_alpha  = out_hidden + (long)n_node * 64;     // n_edge
  float* out_msg    = out_alpha  + (long)n_edge;          // n_edge*64
  float* out_obj    = out_msg    + (long)n_edge * 64;     // n_edge
  float* out_atemp  = out_obj    + (long)n_edge;          // n_edge

  float* ws    = (float*)d_ws;
  float* agg   = ws;                                      // n_node*64
  float* hproj = agg   + (long)n_node * 64;               // n_node*64
  float* rproj = hproj + (long)n_node * 64;               // n_rel*64
  float* qattn = rproj + (long)n_rel  * 64;               // n_query*64

  // 1) clear segment-sum accumulator
  zero_f32<<<1024, 256, 0, stream>>>(agg, (long)n_node * 64);

  // 2) hoisted GEMMs (WMMA f32 16x16x4); 8 waves/block, 16 rows/wave
  gemm64_wmma<<<(n_node  + 127) / 128, 256, 0, stream>>>(hidden, nullptr, Ws,  nullptr, hproj, n_node);
  gemm64_wmma<<<(n_rel   + 127) / 128, 256, 0, stream>>>(rela,   nullptr, Wr,  nullptr, rproj, n_rel);
  gemm64_wmma<<<(n_query + 127) / 128, 256, 0, stream>>>(rela,   q_rel,   Wqr, Wqr_b,   qattn, n_query);

  // 3) memory-bound edge pass: 1 wave / edge, 8 edges / block
  edge_kernel<<<(n_edge + 7) / 8, 256, 0, stream>>>(edges, hidden, rela, hproj, rproj, qattn,
                                                    waw, wab, out_alpha, out_msg, out_obj,
                                                    out_atemp, agg, n_edge);

  // 4) hidden_new = agg @ W_h (WMMA)
  gemm64_wmma<<<(n_node + 127) / 128, 256, 0, stream>>>(agg, nullptr, W_h, nullptr, out_hidden, n_node);
}